// SigJoinLayer_5394478924087
// MI455X (gfx1250) — compile-verified
//
#include <hip/hip_runtime.h>

typedef __attribute__((ext_vector_type(2))) float v2f;
typedef __attribute__((ext_vector_type(8))) float v8f;

#define DPATH     8
#define SIGLEN    299592
#define WPB       5          // workgroups cooperating per batch row
#define NTHREADS  256
#define NTILES    1170       // (SIGLEN - 72) / 256, covers levels 3..6 contiguously

// Level offsets in the flat signature: L1@0(8) L2@8(64) L3@72(512) L4@584(4096)
// L5@4680(32768) L6@37448(262144)

__global__ __launch_bounds__(NTHREADS)
void sigjoin_wmma_kernel(const float* __restrict__ X,
                         const float* __restrict__ Y,
                         float* __restrict__ OUT)
{
    const int b   = blockIdx.x / WPB;
    const int wgr = blockIdx.x % WPB;
    const float* x  = X  + (size_t)b * SIGLEN;
    const float* y  = Y  + (size_t)b * DPATH;
    float*      out = OUT + (size_t)b * SIGLEN;

    __shared__ float ly[8];
    __shared__ float u2[8];        // z2 = x2 + u2 (x) y
    __shared__ float u3[64];
    __shared__ float u4[512];
    __shared__ float u5[4096];
    __shared__ float v5[4096];     // u6[p] = x5[p] + v5[p>>3]*y[p&7]
    __shared__ float s8[8];
    __shared__ float s64[64];
    __shared__ float s512[512];

    const int tid = threadIdx.x;

    if (tid < 8) ly[tid] = y[tid];
    __syncthreads();

    // ---------- Horner chains: g_m^(t) = x_m/t! + g_{m-1}^(t+1) (x) y ----------
    if (tid < 8)  u2[tid] = x[tid] + 0.5f * ly[tid];
    if (tid < 8)  s8[tid] = x[tid] * (1.f/2.f) + ly[tid] * (1.f/6.f);
    __syncthreads();
    if (tid < 64) u3[tid] = x[8 + tid] + s8[tid >> 3] * ly[tid & 7];
    __syncthreads();
    // k=4
    if (tid < 8)  s8[tid] = x[tid] * (1.f/6.f) + ly[tid] * (1.f/24.f);
    __syncthreads();
    if (tid < 64) s64[tid] = x[8 + tid] * 0.5f + s8[tid >> 3] * ly[tid & 7];
    __syncthreads();
    for (int i = tid; i < 512; i += NTHREADS)
        u4[i] = x[72 + i] + s64[i >> 3] * ly[i & 7];
    __syncthreads();
    // k=5
    if (tid < 8)  s8[tid] = x[tid] * (1.f/24.f) + ly[tid] * (1.f/120.f);
    __syncthreads();
    if (tid < 64) s64[tid] = x[8 + tid] * (1.f/6.f) + s8[tid >> 3] * ly[tid & 7];
    __syncthreads();
    for (int i = tid; i < 512; i += NTHREADS)
        s512[i] = x[72 + i] * 0.5f + s64[i >> 3] * ly[i & 7];
    __syncthreads();
    for (int i = tid; i < 4096; i += NTHREADS)
        u5[i] = x[584 + i] + s512[i >> 3] * ly[i & 7];
    __syncthreads();
    // k=6 (stop at g4 = v5; u6 formed on the fly from x5)
    if (tid < 8)  s8[tid] = x[tid] * (1.f/120.f) + ly[tid] * (1.f/720.f);
    __syncthreads();
    if (tid < 64) s64[tid] = x[8 + tid] * (1.f/24.f) + s8[tid >> 3] * ly[tid & 7];
    __syncthreads();
    for (int i = tid; i < 512; i += NTHREADS)
        s512[i] = x[72 + i] * (1.f/6.f) + s64[i >> 3] * ly[i & 7];
    __syncthreads();
    for (int i = tid; i < 4096; i += NTHREADS)
        v5[i] = x[584 + i] * 0.5f + s512[i >> 3] * ly[i & 7];
    __syncthreads();

    // ---------- levels 1 & 2 (tiny, scalar; one WG per batch row) ----------
    if (wgr == 0) {
        if (tid < 8)  out[tid]     = x[tid]     + ly[tid];
        if (tid < 64) out[8 + tid] = x[8 + tid] + u2[tid >> 3] * ly[tid & 7];
    }

    // ---------- levels 3..6 via f32 WMMA rank-1 tiles ----------
    const int lane = tid & 31;
    const int hi   = lane >> 4;    // 0: M=0..15 / K=0,1 half ; 1: M+8 / K=2,3 half
    const int lo   = lane & 15;
    const int coff = 16 * 0 + 128 * hi + lo;   // per-lane C/D base offset

    // B (4x16): row0 = [y0..y7, 0..0], row1 = [0..0, y0..y7], rows 2,3 = 0
    v2f B;
    B[0] = (hi == 0 && lo < 8)  ? ly[lo]     : 0.0f;
    B[1] = (hi == 0 && lo >= 8) ? ly[lo - 8] : 0.0f;

    // Force the tile index into an SGPR so all control flow around the WMMA is
    // scalar (s_cbranch) and EXEC is structurally all-ones at the WMMA.
    const int wslot = __builtin_amdgcn_readfirstlane((int)(threadIdx.x >> 5));
    const int slot  = wslot + 8 * wgr;

    for (int tile = slot; tile < NTILES; tile += 8 * WPB) {
        // tile -> (level, tile-within-level); scalar branches
        int lvl, t;
        if      (tile < 2)   { lvl = 3; t = tile;       }
        else if (tile < 18)  { lvl = 4; t = tile - 2;   }
        else if (tile < 146) { lvl = 5; t = tile - 18;  }
        else                 { lvl = 6; t = tile - 146; }

        const int base = 72 + 256 * tile;           // contiguous over all tiles
        const float* __restrict__ cx = x   + base;
        float*       __restrict__ cz = out + base;

        // A (16x4): A[M,0]=u[32t+2M], A[M,1]=u[32t+2M+1]; K=2,3 lanes zeroed
        float ua0 = 0.0f, ua1 = 0.0f;
        if (lvl == 6) {                               // scalar branch
            if (hi == 0) {                            // lane-divergent loads only
                const int p = 32 * t + 2 * lo;        // even
                const float v = v5[p >> 3];
                const int  q = p & 7;
                ua0 = x[4680 + p]     + v * ly[q];
                ua1 = x[4680 + p + 1] + v * ly[q + 1];
            }
        } else {
            const float* U = (lvl == 3) ? u3 : (lvl == 4) ? u4 : u5;
            if (hi == 0) {
                const int p = 32 * t + 2 * lo;
                ua0 = U[p];
                ua1 = U[p + 1];
            }
        }
        v2f A; A[0] = ua0; A[1] = ua1;

        // C = x tile (lane L, vgpr j <-> flat 16*j + 128*hi + lo)
        v8f c;
#pragma unroll
        for (int j = 0; j < 8; ++j)
            c[j] = cx[16 * j + coff];

        // prefetch the next tile this wave will touch (speculative, RT)
        __builtin_prefetch(cx + 256 * 8 * WPB, 0, 1);

        // D = A*B + C  (full f32; rank-1 update z = x + u (x) y)
        v8f d = __builtin_amdgcn_wmma_f32_16x16x4_f32(
            /*neg_a=*/false, A, /*neg_b=*/false, B,
            /*c_mod=*/(short)0, c, /*reuse_a=*/false, /*reuse_b=*/false);

#pragma unroll
        for (int j = 0; j < 8; ++j)
            cz[16 * j + coff] = d[j];
    }
}

extern "C" void kernel_launch(void* const* d_in, const int* in_sizes, int n_in,
                              void* d_out, int out_size, void* d_ws, size_t ws_size,
                              hipStream_t stream)
{
    const float* x = (const float*)d_in[0];
    const float* y = (const float*)d_in[1];
    float* out = (float*)d_out;
    const int batch = in_sizes[1] / DPATH;   // y is (B, 8)
    dim3 grid(batch * WPB);
    sigjoin_wmma_kernel<<<grid, NTHREADS, 0, stream>>>(x, y, out);
}